// AttentionDecoder_31473520345572
// MI455X (gfx1250) — compile-verified
//
#include <hip/hip_runtime.h>
#include <hip/hip_bf16.h>
#include <math.h>

// Problem constants (B=4, L=128, C=1024, D=512)
#define BB 4
#define LL 128
#define CC 1024
#define DD 512

typedef __attribute__((ext_vector_type(2))) float v2f;
typedef __attribute__((ext_vector_type(4))) float v4f;
typedef __attribute__((ext_vector_type(8))) float v8f;

// CDNA5 hardware tanh (TRANS op) — confirmed single-instruction lowering.
#if defined(__has_builtin)
#if __has_builtin(__builtin_amdgcn_tanhf)
#define FAST_TANH(x) __builtin_amdgcn_tanhf(x)
#endif
#endif
#ifndef FAST_TANH
#define FAST_TANH(x) tanhf(x)
#endif

// ---------------------------------------------------------------------------
// Kernel A: wq[m][e] = sum_d hidden[m][d] * W[e][d]
//   M = B*L = 512, N = D = 512, K = 512. One wave per 16x16 tile,
//   V_WMMA_F32_16X16X4_F32 stepping K by 4 (f32 in/out keeps fp32 accuracy).
// ---------------------------------------------------------------------------
__global__ void __launch_bounds__(32)
wq_gemm_kernel(const float* __restrict__ H, const float* __restrict__ W,
               float* __restrict__ out) {
    const int lane = threadIdx.x;      // full wave, EXEC all-1s
    const int half = lane >> 4;
    const int idx  = lane & 15;
    const int tm = blockIdx.y * 16;    // rows: m over B*L
    const int tn = blockIdx.x * 16;    // cols: e over D

    const float* arow = H + (size_t)(tm + idx) * DD;   // hidden row m
    const float* brow = W + (size_t)(tn + idx) * DD;   // B[k][n] = W[n][k]

    v8f c = {};
#pragma unroll 4
    for (int k = 0; k < DD; k += 4) {
        const int k0 = k + half * 2;
        v2f a = *(const v2f*)(arow + k0);
        v2f b = *(const v2f*)(brow + k0);
        c = __builtin_amdgcn_wmma_f32_16x16x4_f32(false, a, false, b,
                                                  (short)0, c, false, false);
    }
#pragma unroll
    for (int r = 0; r < 8; ++r) {
        out[(size_t)(tm + r + half * 8) * DD + tn + idx] = c[r];
    }
}

// ---------------------------------------------------------------------------
// Kernel B: ucT[b][e][c] = sum_d U[e][d] * ctx[b][c][d] + U_b[e]
//   Computed as U * ctx^T so the result lands TRANSPOSED ([D][C] per batch):
//   the fused stage streams uc with c contiguous (b128 coalesced loads),
//   and the WMMA store here is coalesced too (N-index across lanes).
// ---------------------------------------------------------------------------
__global__ void __launch_bounds__(32)
ucT_gemm_kernel(const float* __restrict__ CTX, const float* __restrict__ U,
                const float* __restrict__ Ub, float* __restrict__ out) {
    const int lane = threadIdx.x;
    const int half = lane >> 4;
    const int idx  = lane & 15;
    const int b  = blockIdx.z;
    const int te = blockIdx.y * 16;    // rows: e over D
    const int tc = blockIdx.x * 16;    // cols: c over C

    const float* arow = U   + (size_t)(te + idx) * DD;
    const float* brow = CTX + ((size_t)b * CC + (tc + idx)) * DD;

    v8f c = {};
#pragma unroll 4
    for (int k = 0; k < DD; k += 4) {
        const int k0 = k + half * 2;
        v2f a  = *(const v2f*)(arow + k0);
        v2f bm = *(const v2f*)(brow + k0);
        c = __builtin_amdgcn_wmma_f32_16x16x4_f32(false, a, false, bm,
                                                  (short)0, c, false, false);
    }
#pragma unroll
    for (int r = 0; r < 8; ++r) {
        const int e = te + r + half * 8;
        out[((size_t)b * DD + e) * CC + tc + idx] = c[r] + Ub[e];
    }
}

// ---------------------------------------------------------------------------
// Kernel C: fused tanh-reduce + softmax.
//   TL=2 l-rows per 256-thread block: each ucT float4 load feeds 2 l-rows
//   (halves L2 traffic to ~512 MB). Thread t owns c = 4t..4t+3 so the per-d
//   uc read is ONE global_load_b128 (coalesced 512B per wave). wq rows + V
//   staged in LDS (broadcast reads). Stable softmax per l-row via LDS
//   reductions; float4 output store.
// ---------------------------------------------------------------------------
#define FUSED_THREADS 256
#define TL 2                                  // l-rows per block (divides 128)
#define C_PER_THREAD  (CC / FUSED_THREADS)    // 4, contiguous

__global__ void __launch_bounds__(FUSED_THREADS)
attn_fused_kernel(const float* __restrict__ wq, const float* __restrict__ ucT,
                  const float* __restrict__ V, float* __restrict__ out) {
    __shared__ float s_wq[TL][DD];
    __shared__ float s_v[DD];
    __shared__ float s_red[FUSED_THREADS];

    const int bl0 = blockIdx.x * TL;     // first (b,l) row of this block
    const int b   = bl0 >> 7;            // / L (L=128; TL divides L so same b)
    const int t   = threadIdx.x;

    // Stage TL wq rows and V into LDS.
    for (int i = t; i < DD; i += FUSED_THREADS) {
#pragma unroll
        for (int il = 0; il < TL; ++il)
            s_wq[il][i] = wq[(size_t)(bl0 + il) * DD + i];
        s_v[i] = V[i];
    }
    __syncthreads();

    const float* ucb = ucT + (size_t)b * DD * CC + 4 * t;  // col base (4 c's)

    float acc[TL][C_PER_THREAD];
#pragma unroll
    for (int il = 0; il < TL; ++il)
#pragma unroll
        for (int j = 0; j < C_PER_THREAD; ++j) acc[il][j] = 0.0f;

#pragma unroll 4
    for (int d = 0; d < DD; ++d) {
        const v4f u  = *(const v4f*)(ucb + (size_t)d * CC);  // one b128 load
        const float vv = s_v[d];
#pragma unroll
        for (int il = 0; il < TL; ++il) {
            const float w = s_wq[il][d];
#pragma unroll
            for (int j = 0; j < C_PER_THREAD; ++j) {
                acc[il][j] += vv * FAST_TANH(w + u[j]);
            }
        }
    }

    // ---- softmax over each C=1024 row ----
#pragma unroll
    for (int il = 0; il < TL; ++il) {
        float lmax = acc[il][0];
#pragma unroll
        for (int j = 1; j < C_PER_THREAD; ++j) lmax = fmaxf(lmax, acc[il][j]);
        __syncthreads();                 // s_red safe to reuse
        s_red[t] = lmax;
        __syncthreads();
        for (int s = FUSED_THREADS / 2; s > 0; s >>= 1) {
            if (t < s) s_red[t] = fmaxf(s_red[t], s_red[t + s]);
            __syncthreads();
        }
        const float rowmax = s_red[0];
        __syncthreads();

        float e[C_PER_THREAD];
        float lsum = 0.0f;
#pragma unroll
        for (int j = 0; j < C_PER_THREAD; ++j) {
            e[j] = __expf(acc[il][j] - rowmax);
            lsum += e[j];
        }
        s_red[t] = lsum;
        __syncthreads();
        for (int s = FUSED_THREADS / 2; s > 0; s >>= 1) {
            if (t < s) s_red[t] += s_red[t + s];
            __syncthreads();
        }
        const float inv = 1.0f / s_red[0];

        v4f o;
#pragma unroll
        for (int j = 0; j < C_PER_THREAD; ++j) o[j] = e[j] * inv;
        *(v4f*)(out + (size_t)(bl0 + il) * CC + 4 * t) = o;   // b128 store
    }
}

// ---------------------------------------------------------------------------
extern "C" void kernel_launch(void* const* d_in, const int* in_sizes, int n_in,
                              void* d_out, int out_size, void* d_ws, size_t ws_size,
                              hipStream_t stream) {
    (void)in_sizes; (void)n_in; (void)out_size; (void)ws_size;

    const float* hidden = (const float*)d_in[0];  // [B,L,D]
    const float* ctx    = (const float*)d_in[1];  // [B,C,D]
    const float* W      = (const float*)d_in[2];  // [D,D]
    const float* U      = (const float*)d_in[3];  // [D,D]
    const float* Ub     = (const float*)d_in[4];  // [D]
    const float* V      = (const float*)d_in[5];  // [D]
    float* out = (float*)d_out;                   // [B,L,C]

    float* ws_wq  = (float*)d_ws;                 // B*L*D floats (1 MB)
    float* ws_ucT = ws_wq + (size_t)BB * LL * DD; // B*D*C floats (8 MB)

    // wq = hidden . W^T   (M = B*L = 512, N = D = 512)
    dim3 gA(DD / 16, (BB * LL) / 16);
    wq_gemm_kernel<<<gA, 32, 0, stream>>>(hidden, W, ws_wq);

    // ucT = (ctx . U^T)^T + b, stored [b][d][c]  (M' = D = 512, N = C = 1024)
    dim3 gB(CC / 16, DD / 16, BB);
    ucT_gemm_kernel<<<gB, 32, 0, stream>>>(ctx, U, Ub, ws_ucT);

    // fused tanh-reduce + softmax, TL l-rows per block
    attn_fused_kernel<<<(BB * LL) / TL, FUSED_THREADS, 0, stream>>>(ws_wq, ws_ucT, V, out);
}